// GAT_41223096107202
// MI455X (gfx1250) — compile-verified
//
#include <hip/hip_runtime.h>
#include <math.h>

typedef __attribute__((ext_vector_type(2))) float v2f;
typedef __attribute__((ext_vector_type(8))) float v8f;

#define HID 256
#define HEADS 8
#define HEAD_DIM 32
#define NEG_SLOPE 0.2f
#define LDS_STRIDE 260   // 256 + 4 pad: row base advances 4 banks/row -> conflict-free

// ---------------------------------------------------------------------------
// GEMM: X[n,256] = IN[n,256] @ W[256,256], fp32 via V_WMMA_F32_16X16X4_F32.
// One 512-thread block = one 16-row M band. Its 16 waves are the 16 N-tiles.
// The shared 16x256 A block is staged once in LDS (cooperative float4 fill),
// then each wave reads A frags via ds_load_b64 and B frags from global (W is
// 256KB, fully L2-resident across the 3125 bands).
// A-frag layout (ISA 16x4 f32): lane L holds M=L%16, K = vgpr + 2*(L/16)
// B-frag layout (4x16):         lane L holds N=L%16, K = vgpr + 2*(L/16)
// C/D layout (16x16 f32):       lane L, vgpr r -> M = r + 8*(L/16), N=L%16
// ---------------------------------------------------------------------------
__global__ __launch_bounds__(512) void gat_gemm_wmma(const float* __restrict__ in,
                                                     const float* __restrict__ W,
                                                     float* __restrict__ X,
                                                     int nrows) {
    __shared__ float Atile[16 * LDS_STRIDE];

    const int m0 = blockIdx.x * 16;
    const int t  = threadIdx.x;              // 0..511

    // Cooperative fill: 512 threads x 8 floats = 16x256 A block.
    {
        const int f   = t * 8;               // flat element index
        const int row = f >> 8;
        const int col = f & 255;
        int grow = m0 + row;
        if (grow >= nrows) grow = nrows - 1; // clamp (duplicate row, unguarded)
        const float4* gp = (const float4*)(in + (size_t)grow * HID + col);
        const float4 u0 = gp[0];
        const float4 u1 = gp[1];
        float* lp = &Atile[row * LDS_STRIDE + col];
        *(float4*)(lp)     = u0;
        *(float4*)(lp + 4) = u1;
    }
    __syncthreads();

    const int nt   = t >> 5;                 // wave id = N-tile 0..15
    const int lane = t & 31;
    const int hi   = lane >> 4;              // 0 or 1
    const int lo   = lane & 15;

    const float* pa = &Atile[lo * LDS_STRIDE];   // A row (LDS) for this lane
    const float* pb = W + nt * 16 + lo;          // B column for this lane

    v8f c0 = {0.f, 0.f, 0.f, 0.f, 0.f, 0.f, 0.f, 0.f};
    v8f c1 = {0.f, 0.f, 0.f, 0.f, 0.f, 0.f, 0.f, 0.f};

#pragma unroll 4
    for (int kb = 0; kb < HID; kb += 8) {
        const int ka = kb + 2 * hi;
        v2f a0 = *(const v2f*)(pa + ka);
        v2f b0;
        b0.x = pb[(size_t)(ka)     * HID];
        b0.y = pb[(size_t)(ka + 1) * HID];
        c0 = __builtin_amdgcn_wmma_f32_16x16x4_f32(false, a0, false, b0,
                                                   (short)0, c0, false, false);
        const int kc = kb + 4 + 2 * hi;
        v2f a1 = *(const v2f*)(pa + kc);
        v2f b1;
        b1.x = pb[(size_t)(kc)     * HID];
        b1.y = pb[(size_t)(kc + 1) * HID];
        c1 = __builtin_amdgcn_wmma_f32_16x16x4_f32(false, a1, false, b1,
                                                   (short)0, c1, false, false);
    }
    const v8f c = c0 + c1;

    float* po = X + nt * 16 + lo;
    if (m0 + 16 <= nrows) {                  // fast path: whole tile valid
#pragma unroll
        for (int r = 0; r < 8; ++r)
            po[(size_t)(m0 + r + 8 * hi) * HID] = c[r];
    } else {
#pragma unroll
        for (int r = 0; r < 8; ++r) {
            const int m = m0 + r + 8 * hi;
            if (m < nrows) po[(size_t)m * HID] = c[r];
        }
    }
}

// ---------------------------------------------------------------------------
// Per-node attention logits: als/ald[n,h] = dot(xh[n,h,:], a_src/a_dst[h,:])
// ---------------------------------------------------------------------------
__global__ void gat_logits(const float* __restrict__ X,
                           const float* __restrict__ a_src,
                           const float* __restrict__ a_dst,
                           float* __restrict__ als,
                           float* __restrict__ ald, int n) {
    const int idx = blockIdx.x * blockDim.x + threadIdx.x;
    if (idx >= n * HEADS) return;
    const int node = idx >> 3, h = idx & 7;
    const float* xp  = X + (size_t)node * HID + h * HEAD_DIM;
    const float* as_ = a_src + h * HEAD_DIM;
    const float* ad_ = a_dst + h * HEAD_DIM;
    float s = 0.f, d = 0.f;
#pragma unroll
    for (int j = 0; j < HEAD_DIM; ++j) {
        const float xv = xp[j];
        s += xv * as_[j];
        d += xv * ad_[j];
    }
    als[idx] = s;
    ald[idx] = d;
}

// out[n,c] = bias[c]  (aggregation then atomic-adds on top)
__global__ void gat_init_out(float* __restrict__ out,
                             const float* __restrict__ b, int n) {
    const int idx = blockIdx.x * blockDim.x + threadIdx.x;
    if (idx >= n * HID) return;
    out[idx] = b[idx & (HID - 1)];
}

__global__ void gat_init_stats(float* __restrict__ mx,
                               float* __restrict__ dn, int n) {
    const int idx = blockIdx.x * blockDim.x + threadIdx.x;
    if (idx >= n * HEADS) return;
    mx[idx] = -INFINITY;
    dn[idx] = 0.f;
}

__device__ inline void edge_sd(const int* __restrict__ ei, int e, int E,
                               int& s, int& d) {
    if (e < E) { s = ei[e]; d = ei[E + e]; }
    else       { s = e - E; d = s; }        // self-loops appended
}

__device__ inline float leaky(float v) {
    return v > 0.f ? v : NEG_SLOPE * v;
}

// Sign-aware float atomic max (init value is -inf).
__device__ inline void atomicMaxF(float* addr, float v) {
    if (v >= 0.f) atomicMax((int*)addr, __float_as_int(v));
    else          atomicMin((unsigned int*)addr, __float_as_uint(v));
}

__global__ void gat_edge_max(const int* __restrict__ ei, int E, int Etot,
                             const float* __restrict__ als,
                             const float* __restrict__ ald,
                             float* __restrict__ mx) {
    const int idx = blockIdx.x * blockDim.x + threadIdx.x;
    if (idx >= Etot * HEADS) return;
    const int e = idx >> 3, h = idx & 7;
    int s, d;
    edge_sd(ei, e, E, s, d);
    const float v = leaky(als[s * HEADS + h] + ald[d * HEADS + h]);
    atomicMaxF(&mx[d * HEADS + h], v);
}

__global__ void gat_edge_denom(const int* __restrict__ ei, int E, int Etot,
                               const float* __restrict__ als,
                               const float* __restrict__ ald,
                               const float* __restrict__ mx,
                               float* __restrict__ dn) {
    const int idx = blockIdx.x * blockDim.x + threadIdx.x;
    if (idx >= Etot * HEADS) return;
    const int e = idx >> 3, h = idx & 7;
    int s, d;
    edge_sd(ei, e, E, s, d);
    const float v = leaky(als[s * HEADS + h] + ald[d * HEADS + h]);
    atomicAdd(&dn[d * HEADS + h], __expf(v - mx[d * HEADS + h]));
}

// One thread handles 8 consecutive channels of one edge (32 threads/edge).
// Channels [t*8, t*8+8) all belong to head t/4. xh gather is two float4
// loads (L2-resident: xh is 51 MB < 192 MB L2), scatter is 8 f32 atomics.
__global__ void gat_edge_aggr(const int* __restrict__ ei, int E, int Etot,
                              const float* __restrict__ als,
                              const float* __restrict__ ald,
                              const float* __restrict__ mx,
                              const float* __restrict__ dn,
                              const float* __restrict__ X,
                              float* __restrict__ out) {
    const unsigned int idx = blockIdx.x * blockDim.x + threadIdx.x;
    if (idx >= (unsigned int)Etot * 32u) return;
    const int e = (int)(idx >> 5);
    const int t = (int)(idx & 31);
    int s, d;
    edge_sd(ei, e, E, s, d);
    const int h = t >> 2;
    const float v = leaky(als[s * HEADS + h] + ald[d * HEADS + h]);
    const float alpha = __expf(v - mx[d * HEADS + h]) / dn[d * HEADS + h];

    const float4* xp = (const float4*)(X + (size_t)s * HID + t * 8);
    const float4 x0 = xp[0];
    const float4 x1 = xp[1];
    float* op = out + (size_t)d * HID + t * 8;
    atomicAdd(op + 0, alpha * x0.x);
    atomicAdd(op + 1, alpha * x0.y);
    atomicAdd(op + 2, alpha * x0.z);
    atomicAdd(op + 3, alpha * x0.w);
    atomicAdd(op + 4, alpha * x1.x);
    atomicAdd(op + 5, alpha * x1.y);
    atomicAdd(op + 6, alpha * x1.z);
    atomicAdd(op + 7, alpha * x1.w);
}

__global__ void gat_elu(float* __restrict__ out, int n) {
    const int idx = blockIdx.x * blockDim.x + threadIdx.x;
    if (idx >= n * HID) return;
    const float v = out[idx];
    out[idx] = v > 0.f ? v : (__expf(v) - 1.f);
}

// ---------------------------------------------------------------------------
extern "C" void kernel_launch(void* const* d_in, const int* in_sizes, int n_in,
                              void* d_out, int out_size, void* d_ws, size_t ws_size,
                              hipStream_t stream) {
    const float* x  = (const float*)d_in[0];
    const int*   ei = (const int*)d_in[1];
    const int n    = in_sizes[0] / HID;   // 50000
    const int E    = in_sizes[1] / 2;     // 800000
    const int Etot = E + n;               // + self loops

    // Workspace carve-up (floats): X (xh), A (ping buffer), 4 small stats.
    float* X   = (float*)d_ws;
    float* A   = X   + (size_t)n * HID;
    float* als = A   + (size_t)n * HID;
    float* ald = als + (size_t)n * HEADS;
    float* mx  = ald + (size_t)n * HEADS;
    float* dn  = mx  + (size_t)n * HEADS;

    const int mtiles = (n + 15) / 16;
    const dim3 blk(256);
    const int g_nh = (n * HEADS + 255) / 256;
    const int g_nc = (n * HID + 255) / 256;
    const int g_eh = (Etot * HEADS + 255) / 256;
    const int g_ec = (int)(((unsigned int)Etot * 32u + 255u) / 256u);

    auto layer = [&](const float* in, const float* W, const float* as_,
                     const float* ad_, const float* b, float* out, bool elu) {
        gat_gemm_wmma<<<mtiles, 512, 0, stream>>>(in, W, X, n);
        gat_logits<<<g_nh, blk, 0, stream>>>(X, as_, ad_, als, ald, n);
        gat_init_out<<<g_nc, blk, 0, stream>>>(out, b, n);
        gat_init_stats<<<g_nh, blk, 0, stream>>>(mx, dn, n);
        gat_edge_max<<<g_eh, blk, 0, stream>>>(ei, E, Etot, als, ald, mx);
        gat_edge_denom<<<g_eh, blk, 0, stream>>>(ei, E, Etot, als, ald, mx, dn);
        gat_edge_aggr<<<g_ec, blk, 0, stream>>>(ei, E, Etot, als, ald, mx, dn, X, out);
        if (elu) gat_elu<<<g_nc, blk, 0, stream>>>(out, n);
    };

    const float* W1 = (const float*)d_in[2],  *as1 = (const float*)d_in[3],
               * ad1 = (const float*)d_in[4], *b1 = (const float*)d_in[5];
    const float* W2 = (const float*)d_in[6],  *as2 = (const float*)d_in[7],
               * ad2 = (const float*)d_in[8], *b2 = (const float*)d_in[9];
    const float* W3 = (const float*)d_in[10], *as3 = (const float*)d_in[11],
               * ad3 = (const float*)d_in[12],*b3 = (const float*)d_in[13];
    const float* W4 = (const float*)d_in[14], *as4 = (const float*)d_in[15],
               * ad4 = (const float*)d_in[16],*b4 = (const float*)d_in[17];

    // Layer k's input buffer can double as its output accumulator: the input
    // is only consumed by the GEMM, which completes (stream order) before
    // init_out overwrites it.
    layer(x, W1, as1, ad1, b1, A, true);
    layer(A, W2, as2, ad2, b2, A, true);
    layer(A, W3, as3, ad3, b3, A, true);
    layer(A, W4, as4, ad4, b4, (float*)d_out, false);
}